// tPCN_4209067950429
// MI455X (gfx1250) — compile-verified
//
#include <hip/hip_runtime.h>
#include <hip/hip_bf16.h>

#define BB 65536
#define H 128
#define NOBS 64
#define NDIR 16
#define MAX_ITER 10
#define TOL 1e-3f
#define LR 1e-3f

#define ROWS 64          // rows per block
#define NT 128           // threads per block (4 waves)

typedef float v2f __attribute__((ext_vector_type(2)));
typedef float v8f __attribute__((ext_vector_type(8)));

// ---- LDS layout (float offsets) ----
#define OFF_WRT   0                      // WrT  K-pair packed [H/2][H][2]    : 16384
#define OFF_WOT   (OFF_WRT + 16384)      // WoutT K-pair packed [H/2][NOBS][2]:  8192
#define OFF_WO    (OFF_WOT + 8192)       // Wout K-pair packed [NOBS/2][H][2] :  8192
#define OFF_PREV  (OFF_WO + 8192)        // [ROWS][H]                          :  8192
#define OFF_CUR   (OFF_PREV + 8192)      // [ROWS][H]                          :  8192
#define OFF_DRV   (OFF_CUR + 8192)       // [ROWS][H]                          :  8192
#define OFF_TMP   (OFF_DRV + 8192)       // [ROWS][H] candidate                :  8192
#define OFF_LJ    (OFF_TMP + 8192)       // [ROWS][NOBS] logits/Jep            :  4096
#define OFF_DIR   (OFF_LJ + 4096)        // int dir idx [ROWS]                 :    64
#define SMEM_FLOATS (OFF_DIR + 64)
#define SMEM_BYTES (SMEM_FLOATS * 4)

// A fragment: 16 rows (lanes 0-15 = M0..15) x 4 K.
// VGPR0: K = k0 + 2*(lane>=16), VGPR1: +1  -> contiguous float2 from row-major LDS.
__device__ __forceinline__ v2f ldA(const float* M, int stride, int wr0, int k0, int lane) {
    int r  = wr0 + (lane & 15);
    int kb = k0 + ((lane >> 4) << 1);
    return *(const v2f*)(M + r * stride + kb);
}

// B fragment from K-pair packed weights Wpk[k>>1][n][2]:
// lane needs B[kb][n], B[kb+1][n] with kb = k0 + 2*(lane>=16) -> contiguous float2.
__device__ __forceinline__ v2f ldB(const float* Wpk, int N, int k0, int n0, int lane) {
    int k2 = (k0 >> 1) + (lane >> 4);
    int n  = n0 + (lane & 15);
    return *(const v2f*)(Wpk + ((size_t)k2 * N + n) * 2);
}

__device__ __forceinline__ v8f wmma4(v2f a, v2f b, v8f c) {
    return __builtin_amdgcn_wmma_f32_16x16x4_f32(false, a, false, b, (short)0, c,
                                                 false, false);
}

__global__ __launch_bounds__(NT)
void tpcn_fused_kernel(const float* __restrict__ prev_state,
                       const float* __restrict__ dir_emb,
                       const float* __restrict__ obs_emb,
                       const float* __restrict__ Wr,
                       const float* __restrict__ Win,
                       const float* __restrict__ Wout,
                       float* __restrict__ out_probs,
                       float* __restrict__ out_cur) {
    extern __shared__ float smem[];
    float* sWrT  = smem + OFF_WRT;
    float* sWoT  = smem + OFF_WOT;
    float* sWo   = smem + OFF_WO;
    float* sPrev = smem + OFF_PREV;
    float* sCur  = smem + OFF_CUR;
    float* sDrv  = smem + OFF_DRV;
    float* sTmp  = smem + OFF_TMP;
    float* sLJ   = smem + OFF_LJ;
    int*   sDir  = (int*)(smem + OFF_DIR);

    const int tid  = threadIdx.x;
    const int lane = tid & 31;
    const int wave = tid >> 5;
    const int wr0  = wave * 16;                 // wave's first local row
    const int row0 = blockIdx.x * ROWS;         // block's first global row

    // ---------- stage weights into LDS (K-pair packed for WMMA B fragments) ----------
    for (int i = tid; i < (H / 2) * H; i += NT) {         // B[k][h] = Wr[h][k]
        int k2 = i / H, h = i % H;
        sWrT[i * 2 + 0] = Wr[h * H + 2 * k2 + 0];
        sWrT[i * 2 + 1] = Wr[h * H + 2 * k2 + 1];
    }
    for (int i = tid; i < (H / 2) * NOBS; i += NT) {      // B[k][o] = Wout[o][k]
        int k2 = i / NOBS, o = i % NOBS;
        sWoT[i * 2 + 0] = Wout[o * H + 2 * k2 + 0];
        sWoT[i * 2 + 1] = Wout[o * H + 2 * k2 + 1];
    }
    for (int i = tid; i < (NOBS / 2) * H; i += NT) {      // B[o][h] = Wout[o][h]
        int o2 = i / H, h = i % H;
        sWo[i * 2 + 0] = Wout[(2 * o2 + 0) * H + h];
        sWo[i * 2 + 1] = Wout[(2 * o2 + 1) * H + h];
    }
    // ---------- stage prev_state rows ----------
    for (int i = tid; i < ROWS * H; i += NT)
        sPrev[i] = prev_state[(size_t)row0 * H + i];

    // ---------- per-row one-hot decode (dir in LDS; obs kept in this thread's reg) ----------
    int obsIdx = 0;
    if (tid < ROWS) {
        int di = 0;
        for (int j = 0; j < NDIR; ++j)
            if (dir_emb[(size_t)(row0 + tid) * NDIR + j] > 0.5f) di = j;
        sDir[tid] = di;
        for (int j = 0; j < NOBS; ++j)
            if (obs_emb[(size_t)(row0 + tid) * NOBS + j] > 0.5f) obsIdx = j;
    }
    __syncthreads();
    // drive[r][h] = Win[h][dir_r]
    for (int i = tid; i < ROWS * H; i += NT) {
        int r = i / H, h = i % H;
        sDrv[i] = Win[h * NDIR + sDir[r]];
    }
    __syncthreads();

    // ---------- initial: cur = norm(relu(prev @ WrT + drive)) ----------
    for (int tp = 0; tp < 8; tp += 2) {                   // paired 16-col tiles of H
        v8f g0 = {}, g1 = {};
        for (int k0 = 0; k0 < H; k0 += 4) {
            v2f a  = ldA(sPrev, H, wr0, k0, lane);
            v2f b0 = ldB(sWrT, H, k0, tp * 16, lane);
            v2f b1 = ldB(sWrT, H, k0, tp * 16 + 16, lane);
            g0 = wmma4(a, b0, g0);
            g1 = wmma4(a, b1, g1);
        }
        int rb = wr0 + ((lane >> 4) << 3), c = lane & 15;
#pragma unroll
        for (int j = 0; j < 8; ++j) {
            int rr = rb + j, c0 = tp * 16 + c, c1 = c0 + 16;
            sTmp[rr * H + c0] = fmaxf(g0[j] + sDrv[rr * H + c0], 0.0f);
            sTmp[rr * H + c1] = fmaxf(g1[j] + sDrv[rr * H + c1], 0.0f);
        }
    }
    __syncthreads();
    if (tid < ROWS) {
        float s = 0.0f;
        for (int h = 0; h < H; ++h) { float v = sTmp[tid * H + h]; s += v * v; }
        float inv = 1.0f / (sqrtf(s) + 1e-6f);
        for (int h = 0; h < H; ++h) sCur[tid * H + h] = sTmp[tid * H + h] * inv;
    }
    __syncthreads();

    // ---------- 10 fixed-point iterations, fully LDS-resident ----------
    for (int it = 0; it < MAX_ITER; ++it) {
        // per-row convergence mask (register-local; same thread uses it later)
        int act = 0;
        if (tid < ROWS) {
            float m = 0.0f;
            for (int h = 0; h < H; ++h) m += sPrev[tid * H + h] - sCur[tid * H + h];
            act = (fabsf(m * (1.0f / H)) > TOL) ? 1 : 0;
        }

        // logits = cur @ WoutT  -> sLJ
        for (int tp = 0; tp < 4; tp += 2) {
            v8f l0 = {}, l1 = {};
            for (int k0 = 0; k0 < H; k0 += 4) {
                v2f a  = ldA(sCur, H, wr0, k0, lane);
                v2f b0 = ldB(sWoT, NOBS, k0, tp * 16, lane);
                v2f b1 = ldB(sWoT, NOBS, k0, tp * 16 + 16, lane);
                l0 = wmma4(a, b0, l0);
                l1 = wmma4(a, b1, l1);
            }
            int rb = wr0 + ((lane >> 4) << 3), c = lane & 15;
#pragma unroll
            for (int j = 0; j < 8; ++j) {
                sLJ[(rb + j) * NOBS + tp * 16 + c]      = l0[j];
                sLJ[(rb + j) * NOBS + tp * 16 + 16 + c] = l1[j];
            }
        }
        __syncthreads();

        // row-wise softmax + Jacobian: Jep_o = f_o * ((onehot_o - f_o) - dot)
        if (tid < ROWS) {
            float mx = -1e30f;
            for (int o = 0; o < NOBS; ++o) mx = fmaxf(mx, sLJ[tid * NOBS + o]);
            float s = 0.0f;
            for (int o = 0; o < NOBS; ++o) {
                float e = __expf(sLJ[tid * NOBS + o] - mx);
                sLJ[tid * NOBS + o] = e;
                s += e;
            }
            float inv = 1.0f / s;
            float dot = 0.0f;
            for (int o = 0; o < NOBS; ++o) {
                float f = sLJ[tid * NOBS + o] * inv;
                sLJ[tid * NOBS + o] = f;
                dot += f * (((o == obsIdx) ? 1.0f : 0.0f) - f);
            }
            for (int o = 0; o < NOBS; ++o) {
                float f = sLJ[tid * NOBS + o];
                sLJ[tid * NOBS + o] = f * ((((o == obsIdx) ? 1.0f : 0.0f) - f) - dot);
            }
        }
        __syncthreads();

        // G = prev @ WrT (+drive), U = Jep @ Wout ; candidate -> sTmp
        for (int tp = 0; tp < 8; tp += 2) {
            v8f g0 = {}, g1 = {}, u0 = {}, u1 = {};
            for (int k0 = 0; k0 < H; k0 += 4) {
                v2f a  = ldA(sPrev, H, wr0, k0, lane);
                v2f b0 = ldB(sWrT, H, k0, tp * 16, lane);
                v2f b1 = ldB(sWrT, H, k0, tp * 16 + 16, lane);
                g0 = wmma4(a, b0, g0);
                g1 = wmma4(a, b1, g1);
            }
            for (int k0 = 0; k0 < NOBS; k0 += 4) {
                v2f a  = ldA(sLJ, NOBS, wr0, k0, lane);
                v2f b0 = ldB(sWo, H, k0, tp * 16, lane);
                v2f b1 = ldB(sWo, H, k0, tp * 16 + 16, lane);
                u0 = wmma4(a, b0, u0);
                u1 = wmma4(a, b1, u1);
            }
            int rb = wr0 + ((lane >> 4) << 3), c = lane & 15;
#pragma unroll
            for (int j = 0; j < 8; ++j) {
                int rr = rb + j, c0 = tp * 16 + c, c1 = c0 + 16;
                float cu0 = sCur[rr * H + c0], cu1 = sCur[rr * H + c1];
                float rg0 = fmaxf(g0[j] + sDrv[rr * H + c0], 0.0f);
                float rg1 = fmaxf(g1[j] + sDrv[rr * H + c1], 0.0f);
                sTmp[rr * H + c0] = cu0 + LR * (rg0 - cu0 + u0[j]);
                sTmp[rr * H + c1] = cu1 + LR * (rg1 - cu1 + u1[j]);
            }
        }
        __syncthreads();

        // conditional update: prev' = cur ; cur' = norm(candidate) (active rows only)
        if (tid < ROWS && act) {
            float s = 0.0f;
            for (int h = 0; h < H; ++h) { float v = sTmp[tid * H + h]; s += v * v; }
            float inv = 1.0f / (sqrtf(s) + 1e-6f);
            for (int h = 0; h < H; ++h) {
                float oc = sCur[tid * H + h];
                sPrev[tid * H + h] = oc;
                sCur[tid * H + h]  = sTmp[tid * H + h] * inv;
            }
        }
        __syncthreads();
    }

    // ---------- final: probs = softmax(cur @ WoutT), emit cur ----------
    for (int tp = 0; tp < 4; tp += 2) {
        v8f l0 = {}, l1 = {};
        for (int k0 = 0; k0 < H; k0 += 4) {
            v2f a  = ldA(sCur, H, wr0, k0, lane);
            v2f b0 = ldB(sWoT, NOBS, k0, tp * 16, lane);
            v2f b1 = ldB(sWoT, NOBS, k0, tp * 16 + 16, lane);
            l0 = wmma4(a, b0, l0);
            l1 = wmma4(a, b1, l1);
        }
        int rb = wr0 + ((lane >> 4) << 3), c = lane & 15;
#pragma unroll
        for (int j = 0; j < 8; ++j) {
            sLJ[(rb + j) * NOBS + tp * 16 + c]      = l0[j];
            sLJ[(rb + j) * NOBS + tp * 16 + 16 + c] = l1[j];
        }
    }
    __syncthreads();
    if (tid < ROWS) {
        float mx = -1e30f;
        for (int o = 0; o < NOBS; ++o) mx = fmaxf(mx, sLJ[tid * NOBS + o]);
        float s = 0.0f;
        for (int o = 0; o < NOBS; ++o) {
            float e = __expf(sLJ[tid * NOBS + o] - mx);
            sLJ[tid * NOBS + o] = e;
            s += e;
        }
        float inv = 1.0f / s;
        for (int o = 0; o < NOBS; ++o)
            out_probs[(size_t)(row0 + tid) * NOBS + o] = sLJ[tid * NOBS + o] * inv;
    }
    for (int i = tid; i < ROWS * H; i += NT)
        out_cur[(size_t)row0 * H + i] = sCur[i];
}

extern "C" void kernel_launch(void* const* d_in, const int* in_sizes, int n_in,
                              void* d_out, int out_size, void* d_ws, size_t ws_size,
                              hipStream_t stream) {
    (void)in_sizes; (void)n_in; (void)out_size; (void)d_ws; (void)ws_size;
    const float* prev = (const float*)d_in[0];
    const float* dire = (const float*)d_in[1];
    const float* obse = (const float*)d_in[2];
    const float* Wr   = (const float*)d_in[3];
    const float* Win  = (const float*)d_in[4];
    const float* Wout = (const float*)d_in[5];
    float* out       = (float*)d_out;
    float* out_probs = out;                       // [B, NOBS]
    float* out_cur   = out + (size_t)BB * NOBS;   // [B, H]

    hipFuncSetAttribute(reinterpret_cast<const void*>(tpcn_fused_kernel),
                        hipFuncAttributeMaxDynamicSharedMemorySize, SMEM_BYTES);
    tpcn_fused_kernel<<<BB / ROWS, NT, SMEM_BYTES, stream>>>(
        prev, dire, obse, Wr, Win, Wout, out_probs, out_cur);
}